// SparseDownsample_3169685865337
// MI455X (gfx1250) — compile-verified
//
#include <hip/hip_runtime.h>

// ---------------------------------------------------------------------------
// SparseConv 3x3/s2 downsample on gfx1250:
//   async-gather (GLOBAL_LOAD_ASYNC_TO_LDS_B128, double-buffered)
//   -> f32 WMMA GEMM (V_WMMA_F32_16X16X4_F32)
//   -> atomic scatter-add -> BatchNorm(batch stats) + ReLU
// ---------------------------------------------------------------------------

typedef __attribute__((ext_vector_type(2))) float v2f;
typedef __attribute__((ext_vector_type(8))) float v8f;

#define CIN       64
#define COUT      128
#define TILE_M    64        // rules per chunk
#define CHUNKS    4         // chunks per block (persistent, W stays in LDS)
#define PW        144       // float2 stride per K-pair row (bank-conflict-free)
#define F_STRIDE  68        // float stride per gathered feature row

// grid = (ceil(ceil(R/64)/CHUNKS), 9), block = 128 threads (4 waves).
__global__ void __launch_bounds__(128)
spconv_gemm_scatter(const float* __restrict__ feats,
                    const float* __restrict__ Wg,           // [9,64,128]
                    const int*   __restrict__ in_idx,       // [9,R]
                    const int*   __restrict__ out_idx,      // [9,R]
                    const unsigned char* __restrict__ mask, // [9,R] bool
                    float*       __restrict__ outAcc,       // [n_out,128]
                    int R)
{
    // W packed as K-pairs: Wp[p][c] = {W[2p][c], W[2p+1][c]}  (p = 0..31)
    __shared__ v2f   Wp[32 * PW];
    __shared__ float Flds[2 * TILE_M * F_STRIDE];

    const int t  = threadIdx.x;
    const int k  = blockIdx.y;
    const size_t kR = (size_t)k * R;

    // --- stage W[k] pre-interleaved into LDS -------------------------------
    {
        const float* Wk = Wg + (size_t)k * CIN * COUT;
        for (int i = t; i < 32 * COUT; i += 128) {
            int p = i >> 7, c = i & 127;
            v2f w2;
            w2.x = Wk[(2 * p) * COUT + c];
            w2.y = Wk[(2 * p + 1) * COUT + c];
            Wp[p * PW + c] = w2;
        }
    }

    const int lane = t & 31;
    const int w    = t >> 5;      // wave id: M rows [16w, 16w+16)
    const int lo   = lane & 15;
    const int hi   = lane >> 4;

    // gather assignment: thread t fills row (t>>1), half (t&1) of each chunk
    const int grow = t >> 1;
    const int gj   = (t & 1) * 8;             // first float4 slot in row
    const int chunk0 = blockIdx.x * CHUNKS * TILE_M;

    // --- async-gather one 64-row chunk into LDS buffer `buf` ---------------
    auto issue_gather = [&](int base, int buf) {
        int rule = base + grow;
        int cr   = rule < R ? rule : R - 1;   // clamp; masked rows never scattered
        int fid  = in_idx[kR + cr];
        const float* src = feats + (size_t)fid * CIN + gj * 4;
        unsigned ldsBase = (unsigned)(uintptr_t)
            &Flds[buf * TILE_M * F_STRIDE + grow * F_STRIDE + gj * 4];
#pragma unroll
        for (int jj = 0; jj < 8; ++jj) {
            unsigned           dst = ldsBase + jj * 16;
            unsigned long long ga  = (unsigned long long)(uintptr_t)(src + jj * 4);
            asm volatile("global_load_async_to_lds_b128 %0, %1, off"
                         :: "v"(dst), "v"(ga) : "memory");
        }
    };

    issue_gather(chunk0, 0);        // prologue: chunk 0 -> buffer 0
    __syncthreads();                // covers Wp staging (ds stores)

    for (int c = 0; c < CHUNKS; ++c) {
        const int buf  = c & 1;
        const int base = chunk0 + c * TILE_M;

        if (c + 1 < CHUNKS) {       // uniform branch: EXEC stays full
            issue_gather(base + TILE_M, buf ^ 1);
            asm volatile("s_wait_asynccnt 0x8" ::: "memory"); // drain current buf
        } else {
            asm volatile("s_wait_asynccnt 0x0" ::: "memory");
        }
        __syncthreads();            // all waves' gathers for `buf` visible

        const float* Fb = &Flds[buf * TILE_M * F_STRIDE];

        v8f acc[8];
#pragma unroll
        for (int n = 0; n < 8; ++n) acc[n] = (v8f)(0.0f);

#pragma unroll
        for (int kk = 0; kk < CIN / 4; ++kk) {
            // A fragment: lanes 0-15 K={4kk,4kk+1}; lanes 16-31 K={4kk+2,4kk+3}
            v2f a = *(const v2f*)&Fb[(16 * w + lo) * F_STRIDE + 4 * kk + 2 * hi];
            const int p = 2 * kk + hi;        // K-pair index for this half-wave
#pragma unroll
            for (int n = 0; n < 8; ++n) {
                v2f b = Wp[p * PW + 16 * n + lo];   // single aligned ds_load_b64
                acc[n] = __builtin_amdgcn_wmma_f32_16x16x4_f32(
                    false, a, false, b, (short)0, acc[n], false, false);
            }
        }

        // scatter-add: D element (vgpr r, lane) -> (M = r + 8*hi, N = 16n+lo)
#pragma unroll
        for (int r = 0; r < 8; ++r) {
            int rule = base + 16 * w + r + 8 * hi;
            if (rule < R) {
                size_t ri = kR + rule;
                if (mask[ri]) {
                    int oid = out_idx[ri];
                    float* dst = outAcc + (size_t)oid * COUT + lo;
#pragma unroll
                    for (int n = 0; n < 8; ++n)
                        atomicAdd(dst + 16 * n, acc[n][r]);
                }
            }
        }
        __syncthreads();            // done reading `buf` before it is refilled
    }
}

// Per-channel sum / sum-of-squares. block = 128 (thread == channel).
#define BN_ROWS 256
__global__ void __launch_bounds__(128)
bn_reduce(const float* __restrict__ outAcc, float* __restrict__ sums, int n_out)
{
    const int c    = threadIdx.x;
    const int row0 = blockIdx.x * BN_ROWS;
    float s = 0.f, s2 = 0.f;
    const int rend = min(row0 + BN_ROWS, n_out);
    for (int r = row0; r < rend; ++r) {
        float v = outAcc[(size_t)r * COUT + c];
        s  += v;
        s2 += v * v;
    }
    atomicAdd(&sums[c], s);
    atomicAdd(&sums[COUT + c], s2);
}

// BN (biased batch stats, matching jnp.var) + ReLU, in place.
__global__ void __launch_bounds__(256)
bn_apply(float* __restrict__ outAcc, const float* __restrict__ sums,
         const float* __restrict__ gamma, const float* __restrict__ beta,
         int n_out)
{
    int i = blockIdx.x * blockDim.x + threadIdx.x;
    int total = n_out * COUT;
    if (i >= total) return;
    int c = i & (COUT - 1);
    float inv_n = 1.0f / (float)n_out;
    float mean  = sums[c] * inv_n;
    float var   = sums[COUT + c] * inv_n - mean * mean;
    float x     = outAcc[i];
    float y     = gamma[c] * (x - mean) * rsqrtf(var + 1e-5f) + beta[c];
    outAcc[i]   = fmaxf(y, 0.0f);
}

extern "C" void kernel_launch(void* const* d_in, const int* in_sizes, int n_in,
                              void* d_out, int out_size, void* d_ws, size_t ws_size,
                              hipStream_t stream)
{
    const float*         feats   = (const float*)d_in[0];
    const float*         Wg      = (const float*)d_in[1];
    const float*         gamma   = (const float*)d_in[2];
    const float*         beta    = (const float*)d_in[3];
    const int*           in_idx  = (const int*)d_in[4];
    const int*           out_idx = (const int*)d_in[5];
    const unsigned char* mask    = (const unsigned char*)d_in[6];
    // d_in[7] = n_out scalar (device); host recovers it from out_size.

    float* outAcc = (float*)d_out;
    float* sums   = (float*)d_ws;     // 256 floats: [sum(128) | sumsq(128)]

    const int R     = in_sizes[4] / 9;
    const int n_out = out_size / COUT;
    if (R <= 0 || n_out <= 0) return;

    hipMemsetAsync(d_out, 0, (size_t)out_size * sizeof(float), stream);
    hipMemsetAsync(d_ws, 0, 2 * COUT * sizeof(float), stream);

    int mchunks = (R + TILE_M - 1) / TILE_M;
    int gx      = (mchunks + CHUNKS - 1) / CHUNKS;
    dim3 grid(gx, 9, 1);
    spconv_gemm_scatter<<<grid, 128, 0, stream>>>(feats, Wg, in_idx, out_idx,
                                                  mask, outAcc, R);

    int rblocks = (n_out + BN_ROWS - 1) / BN_ROWS;
    bn_reduce<<<rblocks, 128, 0, stream>>>(outAcc, sums, n_out);

    int total   = n_out * COUT;
    int ablocks = (total + 255) / 256;
    bn_apply<<<ablocks, 256, 0, stream>>>(outAcc, sums, gamma, beta, n_out);
}